// MultiClass_5815385719218
// MI455X (gfx1250) — compile-verified
//
#include <hip/hip_runtime.h>
#include <math.h>

// ---------------------------------------------------------------------------
// PNA 2-layer GNN forward for MI455X (gfx1250, wave32).
// Edge einsum collapsed across towers into (E x 15) @ (15 x 25) GEMM,
// executed with V_WMMA_F32_16X16X4_F32 on 16-edge tiles (full f32 precision).
// Aggregation (sum / sum^2 / min / max / count) via native L2 float atomics
// (GLOBAL_ATOMIC_ADD_F32 / MIN_NUM_F32 / MAX_NUM_F32).
// ---------------------------------------------------------------------------

typedef float v2f __attribute__((ext_vector_type(2)));
typedef float v8f __attribute__((ext_vector_type(8)));

#define FDIM   5
#define TDIM   5
#define COLS   25        // T*F fused output columns of the edge GEMM
#define KDIM   15        // 3*F
#define AVG_LOG_F 1.6633595878898281f
#define EPS_STD 1e-5f
#define EPS_BN  1e-5f

// Native CDNA5 float min/max atomics (GLOBAL_ATOMIC_MIN_NUM_F32 / MAX_NUM_F32):
// straight-line, no sign-split exec-mask branching.
__device__ __forceinline__ void atomicMinF(float* addr, float v) {
  __hip_atomic_fetch_min(addr, v, __ATOMIC_RELAXED, __HIP_MEMORY_SCOPE_AGENT);
}
__device__ __forceinline__ void atomicMaxF(float* addr, float v) {
  __hip_atomic_fetch_max(addr, v, __ATOMIC_RELAXED, __HIP_MEMORY_SCOPE_AGENT);
}

// ---------------------------------------------------------------------------
// Init per-layer accumulators: S=Q=0, MN=+inf, MX=-inf, cnt=0, bn sums=0
// ---------------------------------------------------------------------------
__global__ __launch_bounds__(256) void init_accum_kernel(
    float* __restrict__ S, float* __restrict__ Q,
    float* __restrict__ MN, float* __restrict__ MX,
    float* __restrict__ cnt, float* __restrict__ bns, int N)
{
  int i = blockIdx.x * 256 + threadIdx.x;
  int total = N * COLS;
  if (i < total) {
    S[i]  = 0.0f;
    Q[i]  = 0.0f;
    MN[i] =  __builtin_inff();
    MX[i] = -__builtin_inff();
  }
  if (i < N)  cnt[i] = 0.0f;
  if (i < 16) bns[i] = 0.0f;
}

__global__ __launch_bounds__(256) void fill_zero_kernel(float* __restrict__ p, int n)
{
  int i = blockIdx.x * 256 + threadIdx.x;
  if (i < n) p[i] = 0.0f;
}

// ---------------------------------------------------------------------------
// Edge kernel: per 16-edge tile, build h = [x[dst], x[src], e] (K=15 pad 16),
// m(16x25) = h @ Wcomb + bpre via 2 column-tiles x 4 chained f32 WMMAs,
// then scatter-aggregate per dst node.
//   A 16x4 frag : lanes 0-15 -> {K=k0,k0+1}, lanes 16-31 -> {K=k0+2,k0+3}
//   B 4x16 frag : same K split, N = lane%16
//   D 16x16     : lane<16 -> rows 0-7 in d[0..7]; lane>=16 -> rows 8-15
// ---------------------------------------------------------------------------
__global__ __launch_bounds__(256) void edge_kernel(
    const float* __restrict__ x,
    const int*   __restrict__ src, const int* __restrict__ dst,
    const float* __restrict__ ea,
    const float* __restrict__ We,  const float* __restrict__ be,
    const float* __restrict__ Wpre, const float* __restrict__ bpre,
    float* __restrict__ S, float* __restrict__ Q,
    float* __restrict__ MN, float* __restrict__ MX,
    float* __restrict__ cnt, int E)
{
  __shared__ float Wc[16][32];        // combined Wpre: [K pad16][col pad32]
  __shared__ float h[8][16][17];      // per-wave 16x16 h tile (pad 17 vs banks)
  __shared__ int   dly[8][16];        // dst node per tile row

  const int tid  = threadIdx.x;
  const int w    = tid >> 5;
  const int lane = tid & 31;

  // Build combined weight matrix: Wc[k][t*5+o] = Wpre[t][k][o]
  for (int i = tid; i < 16 * 32; i += 256) {
    int k = i >> 5, c = i & 31;
    float v = 0.0f;
    if (k < KDIM && c < COLS) v = Wpre[(c / FDIM) * (KDIM * FDIM) + k * FDIM + (c % FDIM)];
    Wc[k][c] = v;
  }

  // Build h tile (one row per edge), lanes 0-15 of each wave
  const int ebase = blockIdx.x * 128 + w * 16;
  if (lane < 16) {
    int e = ebase + lane;
    int d = -1;
    if (e < E) {
      d = dst[e];
      int s = src[e];
      float a = ea[e];
      #pragma unroll
      for (int f = 0; f < FDIM; ++f) h[w][lane][f] = x[d * FDIM + f];
      #pragma unroll
      for (int f = 0; f < FDIM; ++f) h[w][lane][FDIM + f] = x[s * FDIM + f];
      #pragma unroll
      for (int f = 0; f < FDIM; ++f) h[w][lane][2 * FDIM + f] = fmaf(a, We[f], be[f]);
      h[w][lane][15] = 0.0f;
      atomicAdd(&cnt[d], 1.0f);
    } else {
      #pragma unroll
      for (int k = 0; k < 16; ++k) h[w][lane][k] = 0.0f;
    }
    dly[w][lane] = d;
  }
  __syncthreads();

  const int row = lane & 15;   // M (edge in tile) for A; N column for B/D
  const int hi  = lane >> 4;   // lane-half selector

  // A fragments for the 4 K-chunks
  v2f afrag[4];
  #pragma unroll
  for (int kc = 0; kc < 4; ++kc) {
    int k0 = kc * 4 + hi * 2;
    afrag[kc].x = h[w][row][k0];
    afrag[kc].y = h[w][row][k0 + 1];
  }

  #pragma unroll
  for (int nt = 0; nt < 2; ++nt) {
    const int col = nt * 16 + row;
    const float bias = (col < COLS) ? bpre[col] : 0.0f;   // bpre flat (T*F)==col
    v8f acc;
    #pragma unroll
    for (int i = 0; i < 8; ++i) acc[i] = bias;

    #pragma unroll
    for (int kc = 0; kc < 4; ++kc) {
      v2f bfrag;
      int k0 = kc * 4 + hi * 2;
      bfrag.x = Wc[k0][col];
      bfrag.y = Wc[k0 + 1][col];
      acc = __builtin_amdgcn_wmma_f32_16x16x4_f32(
          false, afrag[kc], false, bfrag, (short)0, acc, false, false);
    }

    if (col < COLS) {
      #pragma unroll
      for (int i = 0; i < 8; ++i) {
        int r = hi * 8 + i;          // D row = edge index in tile
        int d = dly[w][r];
        if (d >= 0) {
          float v = acc[i];
          size_t o = (size_t)d * COLS + col;
          atomicAdd(&S[o], v);
          atomicAdd(&Q[o], v * v);
          atomicMinF(&MN[o], v);
          atomicMaxF(&MX[o], v);
        }
      }
    }
  }
}

// ---------------------------------------------------------------------------
// Node kernel: PNA scalers + post-tower MLP + tower mix (Wlin) ; BN partials
// ---------------------------------------------------------------------------
__global__ __launch_bounds__(256) void node_kernel(
    const float* __restrict__ x,
    const float* __restrict__ S, const float* __restrict__ Q,
    const float* __restrict__ MN, const float* __restrict__ MX,
    const float* __restrict__ cnt,
    const float* __restrict__ Wpost, const float* __restrict__ bpost,
    const float* __restrict__ Wlin,  const float* __restrict__ blin,
    float* __restrict__ u, float* __restrict__ bns, int N)
{
  __shared__ float red[10];
  if (threadIdx.x < 10) red[threadIdx.x] = 0.0f;
  __syncthreads();

  const int n = blockIdx.x * 256 + threadIdx.x;
  float uf[FDIM] = {0.f, 0.f, 0.f, 0.f, 0.f};

  if (n < N) {
    float c    = cnt[n];
    float deg  = fmaxf(c, 1.0f);
    float logd = logf(deg + 1.0f);
    float s1   = logd / AVG_LOG_F;
    float s2   = AVG_LOG_F / logd;
    bool  has  = c > 0.0f;

    const float* Sp  = S  + (size_t)n * COLS;
    const float* Qp  = Q  + (size_t)n * COLS;
    const float* MNp = MN + (size_t)n * COLS;
    const float* MXp = MX + (size_t)n * COLS;

    float xr[FDIM];
    #pragma unroll
    for (int f = 0; f < FDIM; ++f) xr[f] = x[(size_t)n * FDIM + f];

    float y[TDIM];
    for (int t = 0; t < TDIM; ++t) {
      const float* Wp = Wpost + t * 65;   // Wpost[t][k][0], k in [0,65)
      float acc = bpost[t];
      #pragma unroll
      for (int f = 0; f < FDIM; ++f) acc = fmaf(xr[f], Wp[f], acc);

      float agg[20];
      #pragma unroll
      for (int o = 0; o < FDIM; ++o) {
        int ci = t * FDIM + o;
        float mean  = Sp[ci] / deg;
        float mean2 = Qp[ci] / deg;
        float var   = mean2 - mean * mean;
        if (var < 0.0f) var = 0.0f;
        float sd = sqrtf(var + EPS_STD);
        agg[o]      = mean;
        agg[5 + o]  = has ? MNp[ci] : 0.0f;
        agg[10 + o] = has ? MXp[ci] : 0.0f;
        agg[15 + o] = sd;
      }
      #pragma unroll
      for (int k = 0; k < 20; ++k) {
        acc = fmaf(agg[k],      Wp[5 + k],  acc);
        acc = fmaf(agg[k] * s1, Wp[25 + k], acc);
        acc = fmaf(agg[k] * s2, Wp[45 + k], acc);
      }
      y[t] = acc;
    }

    #pragma unroll
    for (int f = 0; f < FDIM; ++f) {
      float a = blin[f];
      #pragma unroll
      for (int t = 0; t < TDIM; ++t) a = fmaf(y[t], Wlin[t * FDIM + f], a);
      uf[f] = a;
      u[(size_t)n * FDIM + f] = a;
    }
  }

  // Two-level BN statistics reduction (sum, sum of squares per feature)
  #pragma unroll
  for (int f = 0; f < FDIM; ++f) {
    atomicAdd(&red[f],     uf[f]);
    atomicAdd(&red[5 + f], uf[f] * uf[f]);
  }
  __syncthreads();
  if (threadIdx.x < 10) atomicAdd(&bns[threadIdx.x], red[threadIdx.x]);
}

// ---------------------------------------------------------------------------
// BN finalize: scale = gamma*rsqrt(var+eps), shift = beta - mu*scale
// ---------------------------------------------------------------------------
__global__ void bn_finalize_kernel(const float* __restrict__ bns,
                                   const float* __restrict__ gamma,
                                   const float* __restrict__ beta,
                                   float* __restrict__ stats, float invN)
{
  int f = threadIdx.x;
  if (f < FDIM) {
    float mu  = bns[f] * invN;
    float var = bns[5 + f] * invN - mu * mu;
    float rs  = rsqrtf(var + EPS_BN);
    float sc  = gamma[f] * rs;
    stats[f]     = sc;
    stats[5 + f] = beta[f] - mu * sc;
  }
}

__global__ __launch_bounds__(256) void bn_apply_kernel(
    const float* __restrict__ u, const float* __restrict__ stats,
    float* __restrict__ xo, int total)
{
  int i = blockIdx.x * 256 + threadIdx.x;
  if (i < total) {
    int f = i % FDIM;
    float v = fmaf(u[i], stats[f], stats[5 + f]);
    xo[i] = fmaxf(v, 0.0f);
  }
}

// ---------------------------------------------------------------------------
// Graph pooling: segment_sum over batch (batch is sorted -> good locality)
// ---------------------------------------------------------------------------
__global__ __launch_bounds__(256) void pool_kernel(
    const float* __restrict__ x, const int* __restrict__ batch,
    float* __restrict__ pooled, int N)
{
  int n = blockIdx.x * 256 + threadIdx.x;
  if (n < N) {
    int b = batch[n];
    #pragma unroll
    for (int f = 0; f < FDIM; ++f)
      atomicAdd(&pooled[b * FDIM + f], x[(size_t)n * FDIM + f]);
  }
}

// ---------------------------------------------------------------------------
// Final MLP: 5 -> 5 -> 10 -> 3 per graph
// ---------------------------------------------------------------------------
__global__ __launch_bounds__(256) void mlp_kernel(
    const float* __restrict__ pooled,
    const float* __restrict__ W1, const float* __restrict__ b1,
    const float* __restrict__ W2, const float* __restrict__ b2,
    const float* __restrict__ W3, const float* __restrict__ b3,
    float* __restrict__ out, int G)
{
  int g = blockIdx.x * 256 + threadIdx.x;
  if (g >= G) return;
  float p[5], z1[5], z2[10];
  #pragma unroll
  for (int f = 0; f < 5; ++f) p[f] = pooled[g * 5 + f];
  #pragma unroll
  for (int j = 0; j < 5; ++j) {
    float a = b1[j];
    #pragma unroll
    for (int i = 0; i < 5; ++i) a = fmaf(p[i], W1[i * 5 + j], a);
    z1[j] = fmaxf(a, 0.0f);
  }
  #pragma unroll
  for (int j = 0; j < 10; ++j) {
    float a = b2[j];
    #pragma unroll
    for (int i = 0; i < 5; ++i) a = fmaf(z1[i], W2[i * 10 + j], a);
    z2[j] = fmaxf(a, 0.0f);
  }
  #pragma unroll
  for (int j = 0; j < 3; ++j) {
    float a = b3[j];
    #pragma unroll
    for (int i = 0; i < 10; ++i) a = fmaf(z2[i], W3[i * 3 + j], a);
    out[g * 3 + j] = a;
  }
}

// ---------------------------------------------------------------------------
extern "C" void kernel_launch(void* const* d_in, const int* in_sizes, int n_in,
                              void* d_out, int out_size, void* d_ws, size_t ws_size,
                              hipStream_t stream)
{
  const float* x0    = (const float*)d_in[0];
  const int*   ei    = (const int*)  d_in[1];
  const float* ea    = (const float*)d_in[2];
  const int*   batch = (const int*)  d_in[3];
  const float* We    = (const float*)d_in[4];
  const float* be    = (const float*)d_in[5];
  const float* Wpre  = (const float*)d_in[6];
  const float* bpre  = (const float*)d_in[7];
  const float* Wpost = (const float*)d_in[8];
  const float* bpost = (const float*)d_in[9];
  const float* Wlin  = (const float*)d_in[10];
  const float* blin  = (const float*)d_in[11];
  const float* gamma = (const float*)d_in[12];
  const float* beta  = (const float*)d_in[13];
  const float* W1    = (const float*)d_in[14];
  const float* b1    = (const float*)d_in[15];
  const float* W2    = (const float*)d_in[16];
  const float* b2    = (const float*)d_in[17];
  const float* W3    = (const float*)d_in[18];
  const float* b3    = (const float*)d_in[19];

  const int N = in_sizes[0] / FDIM;
  const int E = in_sizes[1] / 2;
  const int G = out_size / 3;
  const int* src = ei;
  const int* dst = ei + E;

  // Workspace layout (floats)
  float* w = (float*)d_ws;
  float* xA     = w;                 w += (size_t)N * FDIM;
  float* xB     = w;                 w += (size_t)N * FDIM;
  float* cnt    = w;                 w += (size_t)N;
  float* S      = w;                 w += (size_t)N * COLS;
  float* Q      = w;                 w += (size_t)N * COLS;
  float* MN     = w;                 w += (size_t)N * COLS;
  float* MX     = w;                 w += (size_t)N * COLS;
  float* u      = w;                 w += (size_t)N * FDIM;
  float* bns    = w;                 w += 16;      // sums: [0..9]
  float* stats  = w;                 w += 16;      // scale[0..4], shift[5..9]
  float* pooled = w;                 w += (size_t)G * FDIM;

  const int nodeBlocks  = (N + 255) / 256;
  const int accumBlocks = (N * COLS + 255) / 256;
  const int edgeBlocks  = (E + 127) / 128;

  const float* xin = x0;
  float* bufs[2] = { xA, xB };

  for (int l = 0; l < 2; ++l) {
    init_accum_kernel<<<accumBlocks, 256, 0, stream>>>(S, Q, MN, MX, cnt, bns, N);
    edge_kernel<<<edgeBlocks, 256, 0, stream>>>(
        xin, src, dst, ea,
        We + l * FDIM, be + l * FDIM,
        Wpre + l * TDIM * KDIM * FDIM, bpre + l * TDIM * FDIM,
        S, Q, MN, MX, cnt, E);
    node_kernel<<<nodeBlocks, 256, 0, stream>>>(
        xin, S, Q, MN, MX, cnt,
        Wpost + l * TDIM * 65, bpost + l * TDIM,
        Wlin + l * FDIM * FDIM, blin + l * FDIM,
        u, bns, N);
    bn_finalize_kernel<<<1, 32, 0, stream>>>(bns, gamma + l * FDIM, beta + l * FDIM,
                                             stats, 1.0f / (float)N);
    bn_apply_kernel<<<(N * FDIM + 255) / 256, 256, 0, stream>>>(u, stats, bufs[l], N * FDIM);
    xin = bufs[l];
  }

  fill_zero_kernel<<<(G * FDIM + 255) / 256, 256, 0, stream>>>(pooled, G * FDIM);
  pool_kernel<<<nodeBlocks, 256, 0, stream>>>(xin, batch, pooled, N);
  mlp_kernel<<<(G + 255) / 256, 256, 0, stream>>>(pooled, W1, b1, W2, b2, W3, b3,
                                                  (float*)d_out, G);
}